// D3TS_18580028522577
// MI455X (gfx1250) — compile-verified
//
#include <hip/hip_runtime.h>
#include <stdint.h>

// ---------------------------------------------------------------------------
// D3-like dispersion energy for MI455X (gfx1250, wave32).
// One wave handles a 16-atom tile x 64 neighbors. Neighbor indices + pair
// mask are staged into LDS with GLOBAL_LOAD_ASYNC_TO_LDS_B128 (ASYNCcnt),
// falling back to non-temporal b128 loads + ds stores if the async builtins
// are unavailable. Pair energies feed V_WMMA_F32_16X16X4_F32 with an
// all-ones B matrix so the f32 accumulator performs the 64-neighbor row-sum
// across 16 chained WMMAs. Per-atom sums are atomically accumulated into the
// per-molecule output (f32 global atomics; mol_idx is sorted -> low
// contention).
// ---------------------------------------------------------------------------

typedef __attribute__((ext_vector_type(2))) float v2f;
typedef __attribute__((ext_vector_type(8))) float v8f;
typedef __attribute__((ext_vector_type(4))) int   v4i;   // native vector for b128 loads

#define WPB   8      // waves per block (256 threads)
#define APW   16     // atoms per wave (WMMA M dimension)
#define NBMAX 64     // neighbors per atom (reference MAX_NB)

#define D3_A1           0.49484001f
#define D3_A2           5.73083694f
#define D3_S6           1.0f
#define D3_S8           0.78981345f
#define D3_BOHR_INV     1.8897261254578281f
#define D3_HALF_HARTREE 13.605693122994f

#if __has_builtin(__builtin_amdgcn_wmma_f32_16x16x4_f32)
#define HAVE_WMMA_F32X4 1
#else
#define HAVE_WMMA_F32X4 0
#endif

#if __has_builtin(__builtin_amdgcn_global_load_async_to_lds_b128) && \
    __has_builtin(__builtin_amdgcn_s_wait_asynccnt)
#define HAVE_ASYNC_LDS 1
#else
#define HAVE_ASYNC_LDS 0
#endif

#if HAVE_ASYNC_LDS
// Builtin signature (from compiler diagnostics): param0 is a v4i pointer in
// the global (AS1) address space; param1 is the LDS-side pointer; the last
// two args (offset, cpol) must be integer constant expressions.
typedef __attribute__((address_space(1))) v4i* gas1_v4i;
typedef __attribute__((address_space(3))) v4i* las3_v4i;

// Value-preserving casts (int->pointer, no addrspacecast needed):
//  - global flat address == AS1 address.
//  - flat LDS pointers carry the LDS byte offset in their low 32 bits
//    (aperture scheme: addr[31:0] is the logical LDS address).
__device__ __forceinline__ gas1_v4i d3_to_global_v4(const void* p)
{
    return (gas1_v4i)(uintptr_t)p;
}
__device__ __forceinline__ las3_v4i d3_to_lds_v4(void* p)
{
    return (las3_v4i)(uint32_t)(uintptr_t)p;
}
#endif

__global__ __launch_bounds__(WPB * 32) void d3ts_zero_kernel(float* __restrict__ out, int n)
{
    int i = blockIdx.x * blockDim.x + threadIdx.x;
    if (i < n) out[i] = 0.0f;
}

__global__ __launch_bounds__(WPB * 32) void d3ts_energy_kernel(
    const float* __restrict__ disp,    // [n_atoms][2] : c6, alpha
    const float* __restrict__ coord,   // [n_atoms][3]
    const float* __restrict__ r4r2,    // [n_elem]
    const int*   __restrict__ numbers, // [n_atoms]
    const int*   __restrict__ nbmat,   // [n_atoms][NBMAX]
    const int*   __restrict__ pmask,   // [n_atoms][NBMAX] (0/1)
    const int*   __restrict__ molidx,  // [n_atoms]
    float*       __restrict__ out,     // [n_mol]
    int n_atoms, int n_elem)
{
    __shared__ __align__(16) int nb_s[WPB][APW * NBMAX]; // neighbor indices
    __shared__ __align__(16) int mk_s[WPB][APW * NBMAX]; // pair mask (0/1)
    __shared__ float rr_s[128];

    const int tid  = threadIdx.x;
    const int lane = tid & 31;
    const int w    = tid >> 5;

    const int wave_id   = blockIdx.x * WPB + w;
    const int atom_base = wave_id * APW;
    const size_t pair_base = (size_t)atom_base * NBMAX;
    const bool full_tile = (atom_base + APW <= n_atoms) && (atom_base < n_atoms);

#if HAVE_ASYNC_LDS
    // Kick off the async DMA of this wave's nbmat/mask tiles into LDS before
    // anything else; it proceeds while we stage r4r2 and hit the barrier.
    if (full_tile) {
        const int eb = lane * 4; // element base for this lane (16B per lane)
        gas1_v4i nbg = d3_to_global_v4(nbmat + pair_base + eb);
        gas1_v4i mkg = d3_to_global_v4(pmask + pair_base + eb);
        las3_v4i nbl = d3_to_lds_v4(&nb_s[w][eb]);
        las3_v4i mkl = d3_to_lds_v4(&mk_s[w][eb]);
        // Offset/cpol must be constant: advance the pointers by 32 v4i
        // (512 B wave-stride) per step instead of using the immediate.
        #pragma unroll
        for (int c = 0; c < (APW * NBMAX) / (32 * 4); ++c) {
            __builtin_amdgcn_global_load_async_to_lds_b128(nbg + c * 32, nbl + c * 32, 0, 0);
            __builtin_amdgcn_global_load_async_to_lds_b128(mkg + c * 32, mkl + c * 32, 0, 0);
        }
    }
#endif

    // Stage the tiny r4r2 table once per block.
    for (int t = tid; t < n_elem && t < 128; t += WPB * 32) rr_s[t] = r4r2[t];
    __syncthreads();

    if (atom_base >= n_atoms) return; // wave-uniform exit: EXEC stays all-1s

    if (full_tile) {
#if HAVE_ASYNC_LDS
        __builtin_amdgcn_s_wait_asynccnt(0); // LDS tiles resident
#else
        #pragma unroll
        for (int c = 0; c < (APW * NBMAX) / (32 * 4); ++c) {
            const int e = (c * 32 + lane) * 4;
            const v4i nb4 = __builtin_nontemporal_load((const v4i*)(nbmat + pair_base + e));
            const v4i mk4 = __builtin_nontemporal_load((const v4i*)(pmask + pair_base + e));
            *(v4i*)(&nb_s[w][e]) = nb4;
            *(v4i*)(&mk_s[w][e]) = mk4;
        }
#endif
    } else { // tail tile (bounds-checked, wave re-converges afterwards)
        const int lim = (n_atoms - atom_base) * NBMAX;
        for (int e = lane; e < APW * NBMAX; e += 32) {
            int nv = 0, mv = 0;
            if (e < lim) {
                nv = nbmat[pair_base + e];
                mv = pmask[pair_base + e];
            }
            nb_s[w][e] = nv;
            mk_s[w][e] = mv;
        }
    }

    // ---- Per-atom (A-matrix row) data for this lane ----
    // WMMA f32 16x16x4 A layout: lanes 0-15 hold K=0,1 (VGPR0,1) for M=lane;
    //                            lanes 16-31 hold K=2,3 for M=lane-16.
    const int mrow = lane & 15;
    const int khi  = (lane >> 4) << 1; // 0 or 2
    int ia = atom_base + mrow;
    if (ia >= n_atoms) ia = n_atoms - 1; // clamp (tail tile); write guarded later

    const float c6_i    = disp[2 * (size_t)ia + 0];
    const float alpha_i = disp[2 * (size_t)ia + 1];
    const float inv_ai  = __builtin_amdgcn_rcpf(alpha_i);
    const float cix = coord[3 * (size_t)ia + 0];
    const float ciy = coord[3 * (size_t)ia + 1];
    const float ciz = coord[3 * (size_t)ia + 2];
    const float rr_i = rr_s[numbers[ia]];

    const int* nbrow = &nb_s[w][mrow * NBMAX];
    const int* mkrow = &mk_s[w][mrow * NBMAX];

#if HAVE_WMMA_F32X4
    v8f acc = {0.f, 0.f, 0.f, 0.f, 0.f, 0.f, 0.f, 0.f};
    const v2f bones = {1.0f, 1.0f}; // all-ones 4x16 B => D columns = row sums
#else
    float accs = 0.0f;
#endif

    #pragma unroll 4
    for (int t = 0; t < NBMAX / 4; ++t) {
        const int k0 = t * 4 + khi;
        float ev0 = 0.f, ev1 = 0.f;
        #pragma unroll
        for (int p = 0; p < 2; ++p) {
            const int   j     = nbrow[k0 + p];
            const float maskf = mkrow[k0 + p] ? 1.0f : 0.0f;

            // L2-resident random gathers (tables ~5 MB << 192 MB L2)
            const float c6_j    = disp[2 * (size_t)j + 0];
            const float alpha_j = disp[2 * (size_t)j + 1];
            const float rr_j    = rr_s[numbers[j]];
            const float dx = cix - coord[3 * (size_t)j + 0];
            const float dy = ciy - coord[3 * (size_t)j + 1];
            const float dz = ciz - coord[3 * (size_t)j + 2];

            const float d  = sqrtf(dx * dx + dy * dy + dz * dz) * D3_BOHR_INV;
            const float d2 = d * d;
            const float d6 = d2 * d2 * d2;
            const float d8 = d6 * d2;

            float denom = c6_i * (alpha_j * inv_ai)
                        + c6_j * (alpha_i * __builtin_amdgcn_rcpf(alpha_j));
            denom = fmaxf(denom, 1e-4f);
            const float c6ij = 2.0f * c6_i * c6_j * __builtin_amdgcn_rcpf(denom);

            const float rrij = 3.0f * rr_i * rr_j;
            const float r0   = D3_A1 * sqrtf(rrij) + D3_A2;
            const float r02  = r0 * r0;
            const float r06  = r02 * r02 * r02;
            const float r08  = r06 * r02;

            const float e = c6ij * (D3_S6 * __builtin_amdgcn_rcpf(d6 + r06)
                                  + D3_S8 * rrij * __builtin_amdgcn_rcpf(d8 + r08));
            const float ee = maskf * e;
            if (p == 0) ev0 = ee; else ev1 = ee;
        }
#if HAVE_WMMA_F32X4
        v2f a; a.x = ev0; a.y = ev1;
        // D = A(16x4 pair energies) x B(ones) + C : accumulates row sums.
        acc = __builtin_amdgcn_wmma_f32_16x16x4_f32(
                  /*neg_a=*/false, a, /*neg_b=*/false, bones,
                  /*c_mod=*/(short)0, acc, /*reuse_a=*/false, /*reuse_b=*/false);
#else
        accs += ev0 + ev1;
#endif
    }

    // ---- Write-back: e_atom -> out[mol] ----
#if HAVE_WMMA_F32X4
    // D layout: lane n (n<16), VGPR r => (M=r, N=n); lane 16+n, VGPR r => (M=8+r, N=n).
    // Columns are identical (B = ones). Lanes 0-7 emit atoms 0-7, lanes 16-23 atoms 8-15.
    const int r = lane & 7;
    float val = acc[0];
    #pragma unroll
    for (int q = 1; q < 8; ++q) val = (r == q) ? acc[q] : val;
    if (lane < 8 || (lane >= 16 && lane < 24)) {
        const int m  = r + ((lane >= 16) ? 8 : 0);
        const int ai = atom_base + m;
        if (ai < n_atoms)
            atomicAdd(&out[molidx[ai]], -D3_HALF_HARTREE * val);
    }
#else
    const float other = __shfl_down(accs, 16, 32);
    if (lane < 16) {
        const int ai = atom_base + lane;
        if (ai < n_atoms)
            atomicAdd(&out[molidx[ai]], -D3_HALF_HARTREE * (accs + other));
    }
#endif
}

extern "C" void kernel_launch(void* const* d_in, const int* in_sizes, int n_in,
                              void* d_out, int out_size, void* d_ws, size_t ws_size,
                              hipStream_t stream)
{
    (void)n_in; (void)d_ws; (void)ws_size;
    const float* disp    = (const float*)d_in[0]; // disp_param [N,2]
    const float* coord   = (const float*)d_in[1]; // coord      [N,3]
    const float* r4r2    = (const float*)d_in[2]; // r4r2       [n_elem]
    const int*   numbers = (const int*)  d_in[3]; // numbers    [N]
    const int*   nbmat   = (const int*)  d_in[4]; // nbmat      [N,64]
    const int*   pmask   = (const int*)  d_in[5]; // pair_mask  [N,64]
    const int*   molidx  = (const int*)  d_in[6]; // mol_idx    [N]
    float*       out     = (float*)d_out;

    const int n_atoms = in_sizes[3];
    const int n_elem  = in_sizes[2];

    d3ts_zero_kernel<<<(out_size + WPB * 32 - 1) / (WPB * 32), WPB * 32, 0, stream>>>(out, out_size);

    const int n_waves = (n_atoms + APW - 1) / APW;
    const int blocks  = (n_waves + WPB - 1) / WPB;
    d3ts_energy_kernel<<<blocks, WPB * 32, 0, stream>>>(
        disp, coord, r4r2, numbers, nbmat, pmask, molidx, out, n_atoms, n_elem);
}